// BahdanauAttention_66958540145296
// MI455X (gfx1250) — compile-verified
//
#include <hip/hip_runtime.h>

// Problem sizes (from reference): B=32, T=2048, DV=1024.
// Outputs concatenated in d_out: context[B,DV] ++ attn[B,T,1] ++ coverage[B,T,1], all f32.
constexpr int Bsz    = 32;
constexpr int Tn     = 2048;
constexpr int DV     = 1024;
constexpr int NCHUNK = 4;           // T split for extra parallelism
constexpr int TC     = Tn / NCHUNK; // 512

typedef __attribute__((ext_vector_type(2))) float v2f; // A/B operand of wmma f32 16x16x4 (2 VGPRs)
typedef __attribute__((ext_vector_type(8))) float v8f; // C/D operand (8 VGPRs)

// Kernel 1: per-(b, 32-col tile, T-chunk) column sums via ones-matrix WMMA.
// grid = Bsz * (DV/256) * NCHUNK = 512 blocks of 256 threads (8 waves).
// Each wave: 32 columns, two independent 16-col WMMA accumulator chains.
__global__ __launch_bounds__(256) void ctx_partial_kernel(
    const float* __restrict__ vals,   // [B, T, DV]
    float* __restrict__ ws)           // [NCHUNK, B, DV] partial sums
{
    const int tid  = threadIdx.x;
    const int lane = tid & 31;
    const int wave = tid >> 5;
    const int half = lane >> 4;   // 0: lanes 0-15, 1: lanes 16-31
    const int n    = lane & 15;   // N index within 16-wide tile

    const int g      = blockIdx.x;
    const int b      = g >> 4;          // 32 batches
    const int rem    = g & 15;
    const int colblk = rem >> 2;        // 4 blocks of 256 columns
    const int chunk  = rem & 3;         // 4 T-chunks
    const int colbase = colblk * 256 + wave * 32;
    const int t0      = chunk * TC;

    // B-operand element mapping (any bijection of 4 T-rows onto (vgpr,half)
    // slots is valid because A is all-ones):
    //   bv.x (VGPR0) <- values[b, t + 2*half + 0, col]
    //   bv.y (VGPR1) <- values[b, t + 2*half + 1, col]
    const float* p0 = vals + ((size_t)b * Tn + t0 + 2 * half) * DV + (colbase + n);
    const float* p1 = p0 + 16; // second 16-column tile

    const v2f a = {1.0f, 1.0f}; // all-ones A matrix -> D rows = column sums of B
    v8f c0 = {};
    v8f c1 = {};

#pragma unroll 4
    for (int it = 0; it < TC / 4; ++it) {
        v2f bv0, bv1;
        bv0.x = p0[0];
        bv0.y = p0[DV];
        bv1.x = p1[0];
        bv1.y = p1[DV];
        // 8 args: (neg_a, A, neg_b, B, c_mod, C, reuse_a, reuse_b)
        c0 = __builtin_amdgcn_wmma_f32_16x16x4_f32(false, a, false, bv0,
                                                   (short)0, c0, false, false);
        c1 = __builtin_amdgcn_wmma_f32_16x16x4_f32(false, a, false, bv1,
                                                   (short)0, c1, false, false);
        p0 += 4 * DV;
        p1 += 4 * DV;
    }

    // D[M, N] identical for every M (ones A). C/D layout: VGPR0 = row M=0 on
    // lanes 0-15 (N=lane) and row M=8 on lanes 16-31 (N=lane-16). So c*[0] on
    // any lane holds the sum for column N = lane%16 of that tile.
    const float v   = (half == 0) ? c0[0] : c1[0];
    const int   col = colbase + half * 16 + n;
    ws[((size_t)chunk * Bsz + b) * DV + col] = v;
}

// Kernel 2: reduce chunk partials into context; fill attn (=1) and coverage (=t).
// grid = B*T / 256 = 256 blocks of 256 threads (idx in [0, 65536)).
__global__ __launch_bounds__(256) void finalize_kernel(
    const float* __restrict__ ws,   // [NCHUNK, B, DV]
    float* __restrict__ context,    // [B, DV]
    float* __restrict__ attn,       // [B, T]
    float* __restrict__ coverage)   // [B, T]
{
    const int idx = blockIdx.x * blockDim.x + threadIdx.x;

    constexpr int BDV = Bsz * DV; // 32768
    if (idx < BDV) {
        float s = ws[idx] + ws[BDV + idx] + ws[2 * BDV + idx] + ws[3 * BDV + idx];
        context[idx] = s;
    }
    // idx spans exactly B*T = 65536
    attn[idx]     = 1.0f;
    coverage[idx] = (float)(idx & (Tn - 1)); // t = idx % T (exclusive cumsum of ones)
}

extern "C" void kernel_launch(void* const* d_in, const int* in_sizes, int n_in,
                              void* d_out, int out_size, void* d_ws, size_t ws_size,
                              hipStream_t stream) {
    // Inputs (setup_inputs order): 0=query, 1=values, 2..=weights (all dead code).
    const float* vals = (const float*)d_in[1];

    float* out      = (float*)d_out;
    float* context  = out;                       // B*DV   = 32768 floats
    float* attn     = out + Bsz * DV;            // B*T    = 65536 floats
    float* coverage = attn + Bsz * Tn;           // B*T    = 65536 floats
    float* ws       = (float*)d_ws;              // NCHUNK*B*DV = 512 KB, overwritten each call

    ctx_partial_kernel<<<Bsz * (DV / 256) * NCHUNK, 256, 0, stream>>>(vals, ws);
    finalize_kernel<<<(Bsz * Tn) / 256, 256, 0, stream>>>(ws, context, attn, coverage);
}